// XGBackpropLayer_12695923327543
// MI455X (gfx1250) — compile-verified
//
#include <hip/hip_runtime.h>
#include <stdint.h>

// Problem constants (from reference): B=50000, F=256, T=1000, D=6, N=127
#define FEATS      256
#define DEPTH      6
#define NODES      127      // 2^(D+1)-1, heap layout; leaves at 63..126
#define ROWS_PER_WG 128     // samples cached in LDS per workgroup
#define THREADS    512      // 16 waves; 4 thread-quarters per sample
#define ROW_STRIDE 260      // 256 + 4 dword pad: float4-aligned, decorrelates banks

typedef __attribute__((ext_vector_type(4))) unsigned int v4u_t;
typedef __attribute__((ext_vector_type(8))) int          v8i_t;
typedef __attribute__((ext_vector_type(4))) int          v4i_t;

__global__ __launch_bounds__(THREADS) void xgb_forest_kernel(
    const float* __restrict__ x,
    const int*   __restrict__ split_idx,
    const float* __restrict__ split_cond,
    const float* __restrict__ base_score,
    float*       __restrict__ out,
    int B, int T)
{
    __shared__ float tile[ROWS_PER_WG * ROW_STRIDE];   // 130 KB padded x tile
    __shared__ float partial[THREADS];                 // 2 KB partial sums

    const int tid  = threadIdx.x;
    const int s0   = blockIdx.x * ROWS_PER_WG;
    const int rows = min(ROWS_PER_WG, B - s0);

    // ---------------------------------------------------------------
    // Stage x tile (rows x 256 f32) into LDS with a 4-dword row pad.
    // Preferred path: Tensor Data Mover (async DMA, TENSORcnt).
    // ---------------------------------------------------------------
#if defined(__AMDGCN__) && __has_builtin(__builtin_amdgcn_tensor_load_to_lds)
    if (tid < 32) {   // one wave issues the DMA
        const uint64_t gaddr = (uint64_t)(uintptr_t)(x + (size_t)s0 * FEATS);
        const uint32_t laddr = (uint32_t)(uintptr_t)(void*)&tile[0];

        // D# group 0 (ISA 8.3): count=1 | lds_addr | global_addr[56:0] | type=2
        v4u_t g0 = {
            1u,
            laddr,
            (uint32_t)gaddr,
            (((uint32_t)(gaddr >> 32)) & 0x01FFFFFFu) | (2u << 30)
        };
        // D# group 1 (ISA 8.4):
        //  dword0: wg_mask=0 | data_size=2 (4B) | pad_enable=1 |
        //          pad_interval=7 (256 dwords)  | pad_amount=3 (4 dwords)
        //  tensor_dim0=256, tensor_dim1=rows, tile_dim0=256, tile_dim1=rows,
        //  tile_dim2=0, tensor_dim0_stride=256, tensor_dim1_stride=0
        v8i_t g1 = {
            (int)((2u << 16) | (1u << 20) | (7u << 22) | (3u << 25)),
            (int)(((uint32_t)FEATS & 0xFFFFu) << 16),   // abar=0 | dim0 lo16
            (int)(((uint32_t)rows  & 0xFFFFu) << 16),   // dim0 hi16=0 | dim1 lo16
            (int)(((uint32_t)FEATS & 0xFFFFu) << 16),   // dim1 hi16=0 | tile_dim0
            (int)((uint32_t)rows & 0xFFFFu),            // tile_dim1 | tile_dim2=0
            FEATS,                                      // dim0_stride lo32
            0,                                          // dim0_stride hi | dim1_stride lo
            0                                           // dim1_stride hi
        };
        v4i_t g2 = {0, 0, 0, 0};                        // group 2 (2D: disabled)
        v4i_t g3 = {0, 0, 0, 0};                        // group 3 (2D: disabled)
        v8i_t g4 = {0, 0, 0, 0, 0, 0, 0, 0};            // extra group (6-arg form)
        __builtin_amdgcn_tensor_load_to_lds(g0, g1, g2, g3, g4, 0);
        __builtin_amdgcn_s_wait_tensorcnt(0);
    }
#else
    // Fallback: cooperative float4 loads
    for (int i = tid; i < rows * (FEATS / 4); i += THREADS) {
        const int r = i >> 6;              // 64 float4 per row
        const int c = (i & 63) << 2;
        const float4 v = *(const float4*)(x + (size_t)(s0 + r) * FEATS + c);
        float* dst = &tile[r * ROW_STRIDE + c];
        dst[0] = v.x; dst[1] = v.y; dst[2] = v.z; dst[3] = v.w;
    }
#endif
    __syncthreads();

    // ---------------------------------------------------------------
    // Traverse: quarter q of 4 threads/sample owns trees [q*T/4, ...).
    // Tree index is wave-uniform -> node loads hit 1-2 L2 lines/level.
    // Unroll x4 trees for 4 independent latency chains per thread.
    // ---------------------------------------------------------------
    const int quarter = tid >> 7;          // 0..3 (uniform per wave)
    const int s_local = tid & 127;
    const int tq = T >> 2;
    const int t0 = quarter * tq;
    const int t1 = (quarter == 3) ? T : t0 + tq;

    float acc = 0.0f;
    if (s_local < rows) {
        const float* __restrict__ xrow = &tile[s_local * ROW_STRIDE];
        int t = t0;
        for (; t + 3 < t1; t += 4) {
            const int*   __restrict__ ip = split_idx  + (size_t)t * NODES;
            const float* __restrict__ cp = split_cond + (size_t)t * NODES;
            {   // prefetch next tree group (clamped to stay in-bounds)
                const int tp = (t + 4 < t1) ? (t + 4) : t;
                __builtin_prefetch(split_idx  + (size_t)tp * NODES, 0, 3);
                __builtin_prefetch(split_cond + (size_t)tp * NODES, 0, 3);
            }
            int n0 = 0, n1 = 0, n2 = 0, n3 = 0;
#pragma unroll
            for (int d = 0; d < DEPTH; ++d) {
                const int   f0 = ip[n0];
                const int   f1 = ip[NODES + n1];
                const int   f2 = ip[2 * NODES + n2];
                const int   f3 = ip[3 * NODES + n3];
                const float c0 = cp[n0];
                const float c1 = cp[NODES + n1];
                const float c2 = cp[2 * NODES + n2];
                const float c3 = cp[3 * NODES + n3];
                n0 = 2 * n0 + 1 + ((xrow[f0] - c0) > 0.0f ? 1 : 0);
                n1 = 2 * n1 + 1 + ((xrow[f1] - c1) > 0.0f ? 1 : 0);
                n2 = 2 * n2 + 1 + ((xrow[f2] - c2) > 0.0f ? 1 : 0);
                n3 = 2 * n3 + 1 + ((xrow[f3] - c3) > 0.0f ? 1 : 0);
            }
            acc += cp[n0] + cp[NODES + n1] + cp[2 * NODES + n2] + cp[3 * NODES + n3];
        }
        for (; t < t1; ++t) {              // tail trees
            const int*   __restrict__ ip = split_idx  + (size_t)t * NODES;
            const float* __restrict__ cp = split_cond + (size_t)t * NODES;
            int n = 0;
#pragma unroll
            for (int d = 0; d < DEPTH; ++d) {
                const int   f = ip[n];
                const float c = cp[n];
                n = 2 * n + 1 + ((xrow[f] - c) > 0.0f ? 1 : 0);
            }
            acc += cp[n];
        }
    }

    // ---------------------------------------------------------------
    // Combine the 4 per-quarter partial sums; add base_score.
    // ---------------------------------------------------------------
    partial[tid] = acc;
    __syncthreads();
    if (tid < ROWS_PER_WG && tid < rows) {
        const float r = partial[tid] + partial[tid + 128] +
                        partial[tid + 256] + partial[tid + 384];
        out[s0 + tid] = r + base_score[0];
    }
}

extern "C" void kernel_launch(void* const* d_in, const int* in_sizes, int n_in,
                              void* d_out, int out_size, void* d_ws, size_t ws_size,
                              hipStream_t stream) {
    const float* x          = (const float*)d_in[0];
    const int*   split_idx  = (const int*)  d_in[1];
    const float* split_cond = (const float*)d_in[2];
    const float* base_score = (const float*)d_in[3];
    float*       out        = (float*)d_out;

    const int B = in_sizes[0] / FEATS;     // 50000
    const int T = in_sizes[1] / NODES;     // 1000

    const int grid = (B + ROWS_PER_WG - 1) / ROWS_PER_WG;
    hipLaunchKernelGGL(xgb_forest_kernel, dim3(grid), dim3(THREADS), 0, stream,
                       x, split_idx, split_cond, base_score, out, B, T);
}